// OrdinalWeightedError_88046829568217
// MI455X (gfx1250) — compile-verified
//
#include <hip/hip_runtime.h>
#include <stdint.h>

namespace {
constexpr int kBS    = 256;            // 8 wave32 per block
constexpr int kVec   = 4;              // elements per thread per tile
constexpr int kTile  = kBS * kVec;     // 1024 elements per tile
constexpr int kDepth = 4;              // async ring depth (4 tiles in flight)
constexpr int kMaxBlocks = 1024;       // partial pairs -> 16 KB of d_ws
}

// ---- CDNA5 async copy-engine path (probe via __has_builtin, fallback = direct loads)
#if defined(__HIP_DEVICE_COMPILE__) &&                                   \
    __has_builtin(__builtin_amdgcn_global_load_async_to_lds_b128) &&     \
    __has_builtin(__builtin_amdgcn_s_wait_asynccnt)
#define OWE_HAVE_ASYNC 1
// Builtin prototype (from clang diagnostic): non-const v4i* in global AS,
// v4i* in LDS AS. Cast chain: drop const -> bitcast -> addrspacecast.
typedef int owe_v4i __attribute__((vector_size(16)));
typedef owe_v4i __attribute__((address_space(1)))* owe_g_v4ip;
typedef owe_v4i __attribute__((address_space(3)))* owe_l_v4ip;
#define OWE_ASYNC_B128(gp, lp)                                           \
  __builtin_amdgcn_global_load_async_to_lds_b128(                        \
      (owe_g_v4ip)(owe_v4i*)(void*)(gp),                                 \
      (owe_l_v4ip)(owe_v4i*)(void*)(lp), 0, 0)
#else
#define OWE_HAVE_ASYNC 0
#endif

// Per-element math. sE is the extended PCNL table in LDS:
//   sE[0] = -0.25 (x=-1 fallback), sE[1..5] = pcnl[0..4],
//   sE[6] = 1.25 (x=5 fallback),   sE[7] = 1.50 (x=6 fallback)
// Because the fallback x/(C-1) = 0.25*x is linear, the same lerp between
// adjacent table entries reproduces the reference exactly for x in [-1, 6).
// sTW[t] = { pcnl[t], weight[t] }.
__device__ __forceinline__ void owe_element(float x, int tgt,
                                            const float* sE, const float2* sTW,
                                            double& accE, double& accW) {
  float f  = floorf(x);
  int   fi = (int)f;
  fi = fi < -1 ? -1 : (fi > 5 ? 5 : fi);
  float pf = sE[fi + 1];               // these two adjacent reads lower to
  float pc = sE[fi + 2];               // one ds_load_2addr_b32
  float ip = pf + (pc - pf) * (x - f);
  float2 tw = sTW[tgt];                // ds_load_b64 gather
  float d    = ip - tw.x;
  float term = d * d * tw.y;
  accE += (double)term;
  accW += (double)tw.y;
}

__device__ __forceinline__ void owe_build_tables(const float* __restrict__ wgt,
                                                 const float* __restrict__ dist,
                                                 float* sE, float2* sTW) {
  if (threadIdx.x == 0) {
    float d0 = dist[0], d1 = dist[1], d2 = dist[2], d3 = dist[3], d4 = dist[4];
    float c0 = d0, c1 = c0 + d1, c2 = c1 + d2, c3 = c2 + d3, c4 = c3 + d4;
    float denom = 2.0f * c4 - d0 - d4;
    float p0 = ((2.0f * c0 - d0) - d0) / denom;   // == 0
    float p1 = ((2.0f * c1 - d1) - d0) / denom;
    float p2 = ((2.0f * c2 - d2) - d0) / denom;
    float p3 = ((2.0f * c3 - d3) - d0) / denom;
    float p4 = ((2.0f * c4 - d4) - d0) / denom;   // == 1
    sE[0] = -0.25f;
    sE[1] = p0; sE[2] = p1; sE[3] = p2; sE[4] = p3; sE[5] = p4;
    sE[6] = 1.25f; sE[7] = 1.50f;
    sTW[0] = make_float2(p0, wgt[0]);
    sTW[1] = make_float2(p1, wgt[1]);
    sTW[2] = make_float2(p2, wgt[2]);
    sTW[3] = make_float2(p3, wgt[3]);
    sTW[4] = make_float2(p4, wgt[4]);
  }
  __syncthreads();
}

__global__ void owe_main(const float* __restrict__ inp,
                         const int*   __restrict__ tgt,
                         const float* __restrict__ wgt,
                         const float* __restrict__ dist,
                         long long n,
                         double* __restrict__ partials) {
  __shared__ __align__(16) float s_in[kDepth][kTile];
  __shared__ __align__(16) int   s_tg[kDepth][kTile];
  __shared__ float  s_E[8];
  __shared__ float2 s_TW[5];
  __shared__ double s_red[kBS / 32][2];

  const int tid = threadIdx.x;
  owe_build_tables(wgt, dist, s_E, s_TW);

  double accE = 0.0, accW = 0.0;
  const long long G  = (long long)gridDim.x;
  const long long t0 = (long long)blockIdx.x;
  long long done;

#if OWE_HAVE_ASYNC
  // Depth-4 async ring. Each lane fetches exactly its own 16B into LDS and
  // reads back only its own region, so the only synchronization needed is
  // the per-wave ASYNCcnt (no barriers in the steady state). Each issue()
  // is 2 instructions per wave (input + target), so k pending tiles
  // correspond to ASYNCcnt == 2k.
  const long long nt = n / kTile;
  done = nt * (long long)kTile;
#pragma unroll
  for (int b = 0; b < kDepth; ++b) {
    long long tp = t0 + (long long)b * G;
    if (tp < nt) {
      OWE_ASYNC_B128(inp + tp * kTile + tid * 4, &s_in[b][tid * 4]);
      OWE_ASYNC_B128(tgt + tp * kTile + tid * 4, &s_tg[b][tid * 4]);
    }
  }
  int buf = 0;
  for (long long t = t0; t < nt; t += G) {
    // Wait until tile t has landed; keep the younger tiles in flight.
    if      (t + 3 * G < nt) __builtin_amdgcn_s_wait_asynccnt(6);
    else if (t + 2 * G < nt) __builtin_amdgcn_s_wait_asynccnt(4);
    else if (t + 1 * G < nt) __builtin_amdgcn_s_wait_asynccnt(2);
    else                     __builtin_amdgcn_s_wait_asynccnt(0);
    float4 x4 = *(const float4*)&s_in[buf][tid * 4];
    int4   g4 = *(const int4*)  &s_tg[buf][tid * 4];
    owe_element(x4.x, g4.x, s_E, s_TW, accE, accW);
    owe_element(x4.y, g4.y, s_E, s_TW, accE, accW);
    owe_element(x4.z, g4.z, s_E, s_TW, accE, accW);
    owe_element(x4.w, g4.w, s_E, s_TW, accE, accW);
    long long tn = t + (long long)kDepth * G;   // refill the buffer just read
    if (tn < nt) {
      OWE_ASYNC_B128(inp + tn * kTile + tid * 4, &s_in[buf][tid * 4]);
      OWE_ASYNC_B128(tgt + tn * kTile + tid * 4, &s_tg[buf][tid * 4]);
    }
    buf = (buf + 1) & (kDepth - 1);
  }
#else
  // Fallback: direct b128 streaming + explicit prefetch of the next stride.
  const long long n4     = n >> 2;
  const long long stride = G * kBS;
  done = n4 << 2;
  const float4* in4 = (const float4*)inp;
  const int4*   tg4 = (const int4*)tgt;
  for (long long i = t0 * kBS + tid; i < n4; i += stride) {
    if (i + stride < n4) {
      __builtin_prefetch(in4 + i + stride, 0, 1);   // global_prefetch_b8
      __builtin_prefetch(tg4 + i + stride, 0, 1);
    }
    float4 x4 = in4[i];
    int4   g4 = tg4[i];
    owe_element(x4.x, g4.x, s_E, s_TW, accE, accW);
    owe_element(x4.y, g4.y, s_E, s_TW, accE, accW);
    owe_element(x4.z, g4.z, s_E, s_TW, accE, accW);
    owe_element(x4.w, g4.w, s_E, s_TW, accE, accW);
  }
  (void)s_in; (void)s_tg;
#endif

  // Ragged tail (none for N = 2^24, but keep it correct for any n).
  for (long long i = done + t0 * kBS + tid; i < n; i += G * kBS)
    owe_element(inp[i], tgt[i], s_E, s_TW, accE, accW);

  // wave32 tree reduction, then 8-wave LDS combine (fixed order: deterministic)
#pragma unroll
  for (int off = 16; off > 0; off >>= 1) {
    accE += __shfl_down(accE, off, 32);
    accW += __shfl_down(accW, off, 32);
  }
  const int wid = tid >> 5, lane = tid & 31;
  if (lane == 0) { s_red[wid][0] = accE; s_red[wid][1] = accW; }
  __syncthreads();
  if (tid == 0) {
    double e = 0.0, w = 0.0;
#pragma unroll
    for (int i = 0; i < kBS / 32; ++i) { e += s_red[i][0]; w += s_red[i][1]; }
    partials[2 * blockIdx.x]     = e;
    partials[2 * blockIdx.x + 1] = w;
  }
}

__global__ void owe_finish(const double* __restrict__ partials, int nblk,
                           float* __restrict__ out) {
  __shared__ double s_red[kBS / 32][2];
  const int tid = threadIdx.x;
  double e = 0.0, w = 0.0;
  for (int i = tid; i < nblk; i += kBS) {
    e += partials[2 * i];
    w += partials[2 * i + 1];
  }
#pragma unroll
  for (int off = 16; off > 0; off >>= 1) {
    e += __shfl_down(e, off, 32);
    w += __shfl_down(w, off, 32);
  }
  const int wid = tid >> 5, lane = tid & 31;
  if (lane == 0) { s_red[wid][0] = e; s_red[wid][1] = w; }
  __syncthreads();
  if (tid == 0) {
    double E = 0.0, W = 0.0;
#pragma unroll
    for (int i = 0; i < kBS / 32; ++i) { E += s_red[i][0]; W += s_red[i][1]; }
    out[0] = (float)(E / W);
  }
}

extern "C" void kernel_launch(void* const* d_in, const int* in_sizes, int n_in,
                              void* d_out, int out_size, void* d_ws, size_t ws_size,
                              hipStream_t stream) {
  (void)n_in; (void)out_size;
  const float* inp  = (const float*)d_in[0];
  const int*   tgt  = (const int*)d_in[1];
  const float* wgt  = (const float*)d_in[2];
  const float* dist = (const float*)d_in[3];
  const long long n = (long long)in_sizes[0];

  int blocks = kMaxBlocks;
  const size_t pairBytes = 2 * sizeof(double);
  if (ws_size < (size_t)blocks * pairBytes) {
    blocks = (int)(ws_size / pairBytes);
    if (blocks < 1) blocks = 1;
  }
  double* partials = (double*)d_ws;
  float*  out      = (float*)d_out;

  owe_main<<<blocks, kBS, 0, stream>>>(inp, tgt, wgt, dist, n, partials);
  owe_finish<<<1, kBS, 0, stream>>>(partials, blocks, out);
}